// Network_34591666602558
// MI455X (gfx1250) — compile-verified
//
#include <hip/hip_runtime.h>

// Problem constants (match reference)
#define N_INPUTS  1024
#define HPL       2048                    // hidden per layer
#define LAYERS    4
#define TOTAL_HIDDEN (HPL * LAYERS)       // 8192
#define NVALS     (N_INPUTS + TOTAL_HIDDEN) // 9216 (36 KB as f32)
#define N_OUT     512
#define HCONN     2048
#define OCONN     4096

typedef __attribute__((ext_vector_type(2))) float v2f;
typedef __attribute__((ext_vector_type(8))) float v8f;
typedef __attribute__((ext_vector_type(4))) float f4;
typedef __attribute__((ext_vector_type(4))) int   i4;

#define AS1 __attribute__((address_space(1)))
#define AS3 __attribute__((address_space(3)))

#if defined(__has_builtin)
#if __has_builtin(__builtin_amdgcn_global_load_async_to_lds_b128)
#define HAVE_ASYNC_LDS 1
#endif
#if __has_builtin(__builtin_amdgcn_s_wait_asynccnt)
#define HAVE_WAIT_ASYNC_BUILTIN 1
#endif
#endif

__device__ __forceinline__ void wait_asynccnt0() {
#if defined(HAVE_WAIT_ASYNC_BUILTIN)
  __builtin_amdgcn_s_wait_asynccnt(0);
#else
  asm volatile("s_wait_asynccnt 0" ::: "memory");
#endif
}

// ---------------------------------------------------------------------------
// values[0:1024] = input, values[1024:9216] = 0  (reference init state)
// ---------------------------------------------------------------------------
__global__ void init_values(const float* __restrict__ inp,
                            float* __restrict__ values) {
  int i = blockIdx.x * blockDim.x + threadIdx.x;
  if (i < NVALS) values[i] = (i < N_INPUTS) ? inp[i] : 0.0f;
}

// Commit layer-k activations into the values table AFTER the whole layer is
// done (preserves "own-layer gathers read zero" semantics, no RAW race).
__global__ void commit_acts(const float* __restrict__ act,
                            float* __restrict__ values, int seg_base) {
  int i = blockIdx.x * blockDim.x + threadIdx.x;
  values[N_INPUTS + seg_base + i] = act[i];
}

// ---------------------------------------------------------------------------
// One block = 16 rows (neurons). 256 threads = 8 waves; wave w owns rows
// 2w, 2w+1 and sweeps the connection axis with coalesced b128 loads
// (ids: int4, weights: float4, mask: 4 bytes as u32), gathering from the
// LDS-staged values table and accumulating a per-lane f32 partial.
// Partials (16 rows x 32 lanes) go to LDS; wave 0 row-reduces them with
// 8x V_WMMA_F32_16X16X4_F32 against an all-ones B (full f32 precision).
// ---------------------------------------------------------------------------
template <int NCONN, bool DO_TANH>
__global__ __launch_bounds__(256)
void layer_kernel(const int* __restrict__ ids,
                  const float* __restrict__ wgt,
                  const unsigned char* __restrict__ cmask,
                  const unsigned char* __restrict__ amask,   // null unless DO_TANH
                  const float* __restrict__ values,
                  float* __restrict__ out) {
  __shared__ float vals[NVALS];         // 36 KB gather table
  __shared__ float pshare[16 * 33];     // padded partials (row stride 33)

  const int tid  = threadIdx.x;
  const int lane = tid & 31;
  const int wave = tid >> 5;

  // Stage values table into LDS.
#if defined(HAVE_ASYNC_LDS)
  {
    const AS1 char* g = (const AS1 char*)values;
    AS3 char* l = (AS3 char*)vals;
#pragma unroll
    for (int t = 0; t < (NVALS * 4) / (256 * 16); ++t) {   // 9 x b128 per thread
      const int off = (t * 256 + tid) * 16;
      __builtin_amdgcn_global_load_async_to_lds_b128(
          (AS1 i4*)(g + off), (AS3 i4*)(l + off), 0, 0);
    }
    wait_asynccnt0();
  }
#else
  {
    const f4* src = (const f4*)values;
    f4* dst = (f4*)vals;
    for (int t = tid; t < NVALS / 4; t += 256) dst[t] = src[t];
  }
#endif
  __syncthreads();

  const int row0 = blockIdx.x * 16;

  for (int rr = 0; rr < 2; ++rr) {
    const int r = 2 * wave + rr;                        // row within block
    const long long base = (long long)(row0 + r) * NCONN;
    const i4* idp = (const i4*)(ids + base);
    const f4* wp  = (const f4*)(wgt + base);
    const unsigned int* mp = (const unsigned int*)(cmask + base);

    float acc = 0.0f;
#pragma unroll 4
    for (int t = 0; t < NCONN / 128; ++t) {
      const int e = t * 32 + lane;                      // vec4 index, coalesced
      i4 id4 = idp[e];
      f4 w4  = wp[e];
      unsigned int m4 = mp[e];
      // bool mask -> exact 0.0f/1.0f via v_cvt_f32_ubyte0..3 (1 VALU op each)
      float m0f = (float)(m4 & 0xffu);
      float m1f = (float)((m4 >> 8) & 0xffu);
      float m2f = (float)((m4 >> 16) & 0xffu);
      float m3f = (float)(m4 >> 24);
      acc = fmaf(vals[id4.x], w4.x * m0f, acc);
      acc = fmaf(vals[id4.y], w4.y * m1f, acc);
      acc = fmaf(vals[id4.z], w4.z * m2f, acc);
      acc = fmaf(vals[id4.w], w4.w * m3f, acc);
    }
    pshare[r * 33 + lane] = acc;
  }
  __syncthreads();

  if (wave == 0) {
    // A-matrix layout for V_WMMA_F32_16X16X4_F32 (32-bit A, 16x4):
    //   lanes 0-15 : VGPR0=K0, VGPR1=K1, M=lane
    //   lanes 16-31: VGPR0=K2, VGPR1=K3, M=lane-16
    const int m    = lane & 15;
    const int koff = (lane >> 4) * 2;
    v8f c = {};
    v2f b; b.x = 1.0f; b.y = 1.0f;                      // all-ones B => row sums
#pragma unroll
    for (int kb = 0; kb < 32; kb += 4) {
      v2f a;
      a.x = pshare[m * 33 + kb + koff];
      a.y = pshare[m * 33 + kb + koff + 1];
      c = __builtin_amdgcn_wmma_f32_16x16x4_f32(
          /*neg_a=*/false, a, /*neg_b=*/false, b,
          /*c_mod=*/(short)0, c, /*reuse_a=*/false, /*reuse_b=*/false);
    }
    // D[m][n] identical across n. Lane 0 holds rows 0-7 (c[0..7]),
    // lane 16 holds rows 8-15.
    if ((lane & 15) == 0) {
      const int mbase = (lane >> 4) * 8;
#pragma unroll
      for (int v = 0; v < 8; ++v) {
        const int row = row0 + mbase + v;
        float pre = c[v];
        float res;
        if (DO_TANH) {
          res = tanhf(pre) * (amask[row] ? 1.0f : 0.0f);
        } else {
          res = pre;
        }
        out[row] = res;
      }
    }
  }
}

// ---------------------------------------------------------------------------
extern "C" void kernel_launch(void* const* d_in, const int* in_sizes, int n_in,
                              void* d_out, int out_size, void* d_ws, size_t ws_size,
                              hipStream_t stream) {
  const float*         inp   = (const float*)d_in[0];
  const int*           h_ids = (const int*)d_in[1];
  const float*         h_w   = (const float*)d_in[2];
  const unsigned char* h_cm  = (const unsigned char*)d_in[3];  // bool (1B)
  const unsigned char* h_am  = (const unsigned char*)d_in[4];  // bool (1B)
  const int*           o_ids = (const int*)d_in[5];
  const float*         o_w   = (const float*)d_in[6];
  const unsigned char* o_cm  = (const unsigned char*)d_in[7];  // bool (1B)
  float* out = (float*)d_out;

  float* values = (float*)d_ws;       // 9216 floats
  float* act    = values + NVALS;     // 2048 floats scratch per layer

  init_values<<<(NVALS + 255) / 256, 256, 0, stream>>>(inp, values);

  for (int k = 0; k < LAYERS; ++k) {
    const long long off = (long long)k * HPL * HCONN;
    layer_kernel<HCONN, true><<<HPL / 16, 256, 0, stream>>>(
        h_ids + off, h_w + off, h_cm + off, h_am + (long long)k * HPL,
        values, act);
    commit_acts<<<HPL / 256, 256, 0, stream>>>(act, values, k * HPL);
  }

  layer_kernel<OCONN, false><<<N_OUT / 16, 256, 0, stream>>>(
      o_ids, o_w, o_cm, nullptr, values, out);
}